// GCN_40192303956067
// MI455X (gfx1250) — compile-verified
//
#include <hip/hip_runtime.h>

typedef float v2f __attribute__((ext_vector_type(2)));
typedef float v8f __attribute__((ext_vector_type(8)));

// ---------------------------------------------------------------- utilities

__global__ void fill_zero_f32(float* __restrict__ p, long long n) {
  long long i = (long long)blockIdx.x * blockDim.x + threadIdx.x;
  if (i < n) p[i] = 0.0f;
}

// Zero-padded repack: src[R x C] -> dst[Rp x Cp]
__global__ void pad_matrix(const float* __restrict__ src, float* __restrict__ dst,
                           int R, int C, int Rp, int Cp) {
  int i = blockIdx.x * blockDim.x + threadIdx.x;
  if (i >= Rp * Cp) return;
  int r = i / Cp, c = i - r * Cp;
  dst[i] = (r < R && c < C) ? src[r * C + c] : 0.0f;
}

__global__ void deg_count(const int* __restrict__ dst, float* __restrict__ deg, int E) {
  int i = blockIdx.x * blockDim.x + threadIdx.x;
  if (i < E) atomicAdd(&deg[dst[i]], 1.0f);
}

// deg -> dinv = rsqrt(deg + 1)   (self-loop included)
__global__ void finalize_dinv(float* __restrict__ dinv, int N) {
  int i = blockIdx.x * blockDim.x + threadIdx.x;
  if (i < N) dinv[i] = rsqrtf(dinv[i] + 1.0f);
}

// Extract first 3 columns of an N x 16 padded matrix into N x 3
__global__ void compact16to3(const float* __restrict__ in, float* __restrict__ out, int N) {
  int i = blockIdx.x * blockDim.x + threadIdx.x;
  if (i >= N * 3) return;
  int n = i / 3, f = i - n * 3;
  out[i] = in[n * 16 + f];
}

// ------------------------------------------------- dense transform via WMMA
// out[N,F] = H[N,K] @ W[K,F], fp32, V_WMMA_F32_16X16X4_F32.
// K % 4 == 0, F % 16 == 0 (inputs pre-padded) -> zero guards, unconditional
// loads (A pair fuses to one global_load_b64).  One wave computes a full
// 16-row x F strip (CT=F/16 tiles), reusing the A fragment across tiles.
// Fragments (ISA 7.12.2): lane = half*16 + m.
//   A 16x4:  v0 = K(half*2), v1 = K(half*2+1), row m
//   B 4x16:  v0 = K(half*2), v1 = K(half*2+1), col m
//   C 16x16: vgpr i -> row i + half*8, col m
template <int K, int F>
__global__ void wmma_transform(const float* __restrict__ H, const float* __restrict__ W,
                               float* __restrict__ out, int N) {
  constexpr int CT = F / 16;
  const int lane = threadIdx.x & 31;
  const int wave = blockIdx.x * (blockDim.x >> 5) + (threadIdx.x >> 5);
  if (wave >= (N >> 4)) return;   // whole-wave uniform exit: EXEC all-1s at WMMA
  const int row0 = wave << 4;
  const int m    = lane & 15;
  const int half = lane >> 4;

  v8f acc[CT];
#pragma unroll
  for (int c = 0; c < CT; ++c) acc[c] = {};

  const float* __restrict__ hrow = H + (long long)(row0 + m) * K;
#pragma unroll
  for (int k0 = 0; k0 < K; k0 += 4) {
    const int ka = k0 + half * 2;
    v2f a;
    a.x = hrow[ka];         // contiguous pair -> b64 load
    a.y = hrow[ka + 1];
#pragma unroll
    for (int c = 0; c < CT; ++c) {
      v2f b;
      b.x = W[ka * F + c * 16 + m];
      b.y = W[(ka + 1) * F + c * 16 + m];
      acc[c] = __builtin_amdgcn_wmma_f32_16x16x4_f32(false, a, false, b,
                                                     (short)0, acc[c], false, false);
    }
  }
#pragma unroll
  for (int c = 0; c < CT; ++c)
#pragma unroll
    for (int i = 0; i < 8; ++i)
      out[(long long)(row0 + i + half * 8) * F + c * 16 + m] = acc[c][i];
}

// ----------------------------------------------------- edge scatter (HBM-bound)
// agg[dst] += dinv[src]*dinv[dst] * h[src], CHUNK features per thread.
template <int F, int CHUNK>
__global__ void edge_agg(const int* __restrict__ src, const int* __restrict__ dst,
                         const float* __restrict__ dinv, const float* __restrict__ h,
                         float* __restrict__ agg, int E) {
  constexpr int CPE = F / CHUNK;                   // chunks per edge (const div)
  const long long idx = (long long)blockIdx.x * blockDim.x + threadIdx.x;
  if (idx >= (long long)E * CPE) return;
  const int e = (int)(idx / CPE);
  const int c = (int)(idx - (long long)e * CPE);
  const int s = src[e];
  const int d = dst[e];
  const float w = dinv[s] * dinv[d];
  const float* hp = h   + (long long)s * F + c * CHUNK;
  float*       ap = agg + (long long)d * F + c * CHUNK;
  if constexpr (CHUNK == 4) {
    const float4 v = *reinterpret_cast<const float4*>(hp);   // 16B L2-resident gather
    atomicAdd(ap + 0, w * v.x);
    atomicAdd(ap + 1, w * v.y);
    atomicAdd(ap + 2, w * v.z);
    atomicAdd(ap + 3, w * v.w);
  } else {
#pragma unroll
    for (int j = 0; j < CHUNK; ++j) atomicAdd(ap + j, w * hp[j]);
  }
}

// ----------------------- fused self-loop + bias (+BN eval) (+ReLU) epilogue
template <int F, bool BN, bool RELU>
__global__ void post_epilogue(const float* __restrict__ hw, const float* __restrict__ agg,
                              const float* __restrict__ dinv, const float* __restrict__ b,
                              const float* __restrict__ g, const float* __restrict__ be,
                              const float* __restrict__ m, const float* __restrict__ v,
                              float* __restrict__ out, int N) {
  const long long i = (long long)blockIdx.x * blockDim.x + threadIdx.x;
  if (i >= (long long)N * F) return;
  const int n = (int)(i / F);
  const int f = (int)(i - (long long)n * F);
  const float di = dinv[n];
  float val = agg[i] + di * di * hw[i] + b[f];
  if constexpr (BN) val = (val - m[f]) * rsqrtf(v[f] + 1e-5f) * g[f] + be[f];
  if constexpr (RELU) val = fmaxf(val, 0.0f);
  out[i] = val;
}

// ---------------------------------------------------------------- launcher

static inline int nblk(long long n, int b) { return (int)((n + b - 1) / b); }

extern "C" void kernel_launch(void* const* d_in, const int* in_sizes, int n_in,
                              void* d_out, int out_size, void* d_ws, size_t ws_size,
                              hipStream_t stream) {
  const float* x   = (const float*)d_in[0];
  const int*   src = (const int*)  d_in[1];
  const int*   dst = (const int*)  d_in[2];
  const float* W1 = (const float*)d_in[3];  const float* b1 = (const float*)d_in[4];
  const float* g1 = (const float*)d_in[5];  const float* be1= (const float*)d_in[6];
  const float* m1 = (const float*)d_in[7];  const float* v1 = (const float*)d_in[8];
  const float* W2 = (const float*)d_in[9];  const float* b2 = (const float*)d_in[10];
  const float* g2 = (const float*)d_in[11]; const float* be2= (const float*)d_in[12];
  const float* m2 = (const float*)d_in[13]; const float* v2 = (const float*)d_in[14];
  const float* W3 = (const float*)d_in[15]; const float* b3 = (const float*)d_in[16];
  const float* W4 = (const float*)d_in[17]; const float* b4 = (const float*)d_in[18];

  const int N = in_sizes[0] / 5;   // 100000 (multiple of 16)
  const int E = in_sizes[1];       // 3200000

  // workspace: dinv[N] | xp/hc[N*8] | hT[N*64] | agg[N*64] | act[N*64] | Wp1[8*64] | Wp4[16*16]
  float* dinv = (float*)d_ws;
  float* xp   = dinv + N;                 // N x 8 padded x; reused as hc (N x 3) for layer 4
  float* hT   = xp  + (size_t)N * 8;
  float* agg  = hT  + (size_t)N * 64;
  float* act  = agg + (size_t)N * 64;
  float* Wp1  = act + (size_t)N * 64;     // 8  x 64
  float* Wp4  = Wp1 + 8 * 64;             // 16 x 16

  // ---- padded repacks (deterministic, every call)
  pad_matrix<<<nblk((long long)N * 8, 256), 256, 0, stream>>>(x,  xp,  N, 5, N, 8);
  pad_matrix<<<nblk(8 * 64,  256), 256, 0, stream>>>(W1, Wp1, 5, 64, 8, 64);
  pad_matrix<<<nblk(16 * 16, 256), 256, 0, stream>>>(W4, Wp4, 16, 3, 16, 16);

  // ---- degrees -> dinv
  fill_zero_f32<<<nblk(N, 256), 256, 0, stream>>>(dinv, N);
  deg_count<<<nblk(E, 256), 256, 0, stream>>>(dst, dinv, E);
  finalize_dinv<<<nblk(N, 256), 256, 0, stream>>>(dinv, N);

  const int tfmGrid = nblk((long long)(N >> 4), 8);   // 8 waves / 256-thread block

  // ---- layer 1: (N x 8) @ (8 x 64), BN + ReLU
  wmma_transform<8, 64><<<tfmGrid, 256, 0, stream>>>(xp, Wp1, hT, N);
  fill_zero_f32<<<nblk((long long)N * 64, 256), 256, 0, stream>>>(agg, (long long)N * 64);
  edge_agg<64, 4><<<nblk((long long)E * 16, 256), 256, 0, stream>>>(src, dst, dinv, hT, agg, E);
  post_epilogue<64, true, true><<<nblk((long long)N * 64, 256), 256, 0, stream>>>(
      hT, agg, dinv, b1, g1, be1, m1, v1, act, N);

  // ---- layer 2: (N x 64) @ (64 x 32), BN + ReLU
  wmma_transform<64, 32><<<tfmGrid, 256, 0, stream>>>(act, W2, hT, N);
  fill_zero_f32<<<nblk((long long)N * 32, 256), 256, 0, stream>>>(agg, (long long)N * 32);
  edge_agg<32, 4><<<nblk((long long)E * 8, 256), 256, 0, stream>>>(src, dst, dinv, hT, agg, E);
  post_epilogue<32, true, true><<<nblk((long long)N * 32, 256), 256, 0, stream>>>(
      hT, agg, dinv, b2, g2, be2, m2, v2, act, N);

  // ---- layer 3: (N x 32) @ (32 x 16), ReLU
  wmma_transform<32, 16><<<tfmGrid, 256, 0, stream>>>(act, W3, hT, N);
  fill_zero_f32<<<nblk((long long)N * 16, 256), 256, 0, stream>>>(agg, (long long)N * 16);
  edge_agg<16, 4><<<nblk((long long)E * 4, 256), 256, 0, stream>>>(src, dst, dinv, hT, agg, E);
  post_epilogue<16, false, true><<<nblk((long long)N * 16, 256), 256, 0, stream>>>(
      hT, agg, dinv, b3, nullptr, nullptr, nullptr, nullptr, act, N);

  // ---- layer 4: (N x 16) @ (16 x 16 padded), compact to width 3, no BN/ReLU
  wmma_transform<16, 16><<<tfmGrid, 256, 0, stream>>>(act, Wp4, hT, N);
  float* hc = xp;                                        // reuse: N x 3 compact transform
  compact16to3<<<nblk((long long)N * 3, 256), 256, 0, stream>>>(hT, hc, N);
  fill_zero_f32<<<nblk((long long)N * 3, 256), 256, 0, stream>>>(agg, (long long)N * 3);
  edge_agg<3, 3><<<nblk((long long)E, 256), 256, 0, stream>>>(src, dst, dinv, hc, agg, E);
  post_epilogue<3, false, false><<<nblk((long long)N * 3, 256), 256, 0, stream>>>(
      hc, agg, dinv, b4, nullptr, nullptr, nullptr, nullptr, (float*)d_out, N);
}